// CrossMerge_34059090657946
// MI455X (gfx1250) — compile-verified
//
#include <hip/hip_runtime.h>
#include <cstdint>
#include <cstddef>

#define B_   16
#define K_   4
#define K2_  2
#define D_   192
#define L_   3136
#define TPB  256

// ---------------------------------------------------------------------------
// Kernel 1: inv[b,k2,:] = argsort(vec_indices[b,:,k2])  (stable)
// Values are in [0, L), so a stable counting sort reproduces jnp.argsort
// (stable=True) exactly: histogram -> exclusive scan -> in-order placement.
// One 256-thread block per (b,k2); 32 blocks total.
// ---------------------------------------------------------------------------
__global__ __launch_bounds__(TPB) void build_inv_kernel(
    const long long* __restrict__ vec_indices,  // (B, L, K2) int64
    int* __restrict__ inv)                      // (B, K2, L) int32
{
    __shared__ int offs[L_];      // histogram, then running exclusive offsets
    __shared__ int partial[TPB];  // per-thread segment sums for the scan
    __shared__ int vals[TPB];     // current chunk of values

    const int bk  = blockIdx.x;   // 0..B*K2-1
    const int b   = bk >> 1;
    const int k2  = bk & 1;
    const int tid = threadIdx.x;
    const long long* src = vec_indices + (size_t)b * L_ * K2_ + k2;

    // 1) zero + histogram
    for (int i = tid; i < L_; i += TPB) offs[i] = 0;
    __syncthreads();
    for (int j = tid; j < L_; j += TPB) {
        int v = (int)src[(size_t)j * K2_];
        if ((unsigned)v >= (unsigned)L_) v = 0;   // defensive
        atomicAdd(&offs[v], 1);
    }
    __syncthreads();

    // 2) exclusive prefix sum over offs[0..L-1] (segmented + Hillis-Steele)
    const int SEG = (L_ + TPB - 1) / TPB;  // 13
    const int s0 = tid * SEG;
    const int s1 = min(s0 + SEG, L_);
    int sum = 0;
    for (int i = s0; i < s1; ++i) sum += offs[i];
    partial[tid] = sum;
    __syncthreads();
    for (int off = 1; off < TPB; off <<= 1) {
        int v = (tid >= off) ? partial[tid - off] : 0;
        __syncthreads();
        partial[tid] += v;
        __syncthreads();
    }
    int run = (tid == 0) ? 0 : partial[tid - 1];
    for (int i = s0; i < s1; ++i) { int c = offs[i]; offs[i] = run; run += c; }
    __syncthreads();

    // 3) stable placement, 256 elements (in index order) per round
    int* invdst = inv + (size_t)bk * L_;
    for (int base = 0; base < L_; base += TPB) {
        const int j = base + tid;
        int v = -1;
        if (j < L_) {
            v = (int)src[(size_t)j * K2_];
            if ((unsigned)v >= (unsigned)L_) v = 0;
        }
        vals[tid] = v;
        __syncthreads();
        int  rank = 0;
        bool last = true;
        if (j < L_) {
            for (int t = 0; t < tid; ++t) rank += (vals[t] == v);
            for (int t = tid + 1; t < TPB; ++t)
                if (vals[t] == v) { last = false; break; }
            invdst[offs[v] + rank] = j;   // inv[pos] = original index (stable)
        }
        __syncthreads();
        if (j < L_ && last) offs[v] += rank + 1;  // one writer per value
        __syncthreads();
    }
}

// ---------------------------------------------------------------------------
// Kernel 2: out[b,d,l] = ys0[inv0[l]] + ys2[L-1-inv0[l]]
//                      + ys1[inv1[l]] + ys3[L-1-inv1[l]]
// One block per (b,d). The four 12.5KB rows are staged into LDS with the
// CDNA5 async global->LDS path (global_load_async_to_lds_b128, ASYNCcnt),
// then the permutation gather is resolved from LDS; outputs are coalesced.
// 50176 B LDS per block -> 6 blocks per 320KB WGP.
// ---------------------------------------------------------------------------
__global__ __launch_bounds__(TPB) void cross_merge_kernel(
    const float* __restrict__ ys,    // (B, K, D, L) f32
    const int*   __restrict__ inv,   // (B, K2, L) int32
    float*       __restrict__ out)   // (B, D, L) f32
{
    __shared__ float ysb[K_ * L_];   // 4 rows * 3136 f32 = 50176 bytes

    const int d   = blockIdx.x;
    const int b   = blockIdx.y;
    const int tid = threadIdx.x;

    // LDS byte offset of the staging buffer (flat-pointer low 32 bits == LDS addr)
    const unsigned lds_base = (unsigned)(uintptr_t)(void*)&ysb[0];

    const int NCHUNK = (L_ * 4) / 16;  // 784 x 16-byte chunks per row
#pragma unroll
    for (int k = 0; k < K_; ++k) {
        const float* rowp = ys + ((((size_t)b) * K_ + k) * D_ + d) * (size_t)L_;
        const unsigned ldsrow = lds_base + (unsigned)(k * L_ * 4);
        for (int c = tid; c < NCHUNK; c += TPB) {
            const unsigned goff  = (unsigned)(c * 16);  // byte offset in row
            const unsigned laddr = ldsrow + goff;       // LDS byte address
            // GVS mode: mem = SADDR(64) + VADDR(32); dest LDS addr in VDST.
            asm volatile("global_load_async_to_lds_b128 %0, %1, %2"
                         :
                         : "v"(laddr), "v"(goff), "s"(rowp)
                         : "memory");
        }
    }
    // Each wave waits for its own async transfers, then the barrier makes
    // all LDS writes visible block-wide.
    asm volatile("s_wait_asynccnt 0" ::: "memory");
    __syncthreads();

    const int*  invp = inv + (size_t)b * (K2_ * L_);
    float*      outp = out + ((size_t)b * D_ + d) * (size_t)L_;
    for (int l = tid; l < L_; l += TPB) {
        const int j0 = invp[l];        // inv[b,0,l]
        const int j1 = invp[L_ + l];   // inv[b,1,l]
        const float v = ysb[j0]
                      + ysb[2 * L_ + (L_ - 1 - j0)]
                      + ysb[L_ + j1]
                      + ysb[3 * L_ + (L_ - 1 - j1)];
        outp[l] = v;
    }
}

// ---------------------------------------------------------------------------
extern "C" void kernel_launch(void* const* d_in, const int* in_sizes, int n_in,
                              void* d_out, int out_size, void* d_ws, size_t ws_size,
                              hipStream_t stream)
{
    const float*     ys          = (const float*)d_in[0];      // (B,K,D,H,W) f32
    const long long* vec_indices = (const long long*)d_in[1];  // (B,L,K2) i64
    float*           out         = (float*)d_out;              // (B,D,L) f32
    int*             inv         = (int*)d_ws;                 // (B,K2,L) i32, 401408 B

    build_inv_kernel<<<dim3(B_ * K2_), dim3(TPB), 0, stream>>>(vec_indices, inv);
    cross_merge_kernel<<<dim3(D_, B_), dim3(TPB), 0, stream>>>(ys, inv, out);

    (void)in_sizes; (void)n_in; (void)out_size; (void)ws_size;
}